// LatticeVariance_20220706030023
// MI455X (gfx1250) — compile-verified
//
#include <hip/hip_runtime.h>
#include <hip/hip_bf16.h>
#include <math.h>

typedef float v2f __attribute__((ext_vector_type(2)));
typedef float v8f __attribute__((ext_vector_type(8)));

#define NTRI     392
#define NTRI_PAD 400
#define NTILE    (NTRI_PAD / 16)      // 25
#define IMG_H    192
#define IMG_W    192
#define NPIX     (IMG_H * IMG_W)      // 36864
#define NB       2
#define TOTPIX   (NB * NPIX)          // 73728
#define NGP      (15 * 15)            // 225 grid points

// Branchless ascending top-3 insertion: pure v_cndmask chains, no EXEC churn.
static __device__ __forceinline__ void top3_insert(float d, int t,
    float& d0, float& d1, float& d2, int& i0, int& i1, int& i2) {
  const bool lt0 = d < d0, lt1 = d < d1, lt2 = d < d2;
  float nd0 = lt0 ? d  : d0;
  int   ni0 = lt0 ? t  : i0;
  float nd1 = lt0 ? d0 : (lt1 ? d : d1);
  int   ni1 = lt0 ? i0 : (lt1 ? t : i1);
  float nd2 = lt1 ? d1 : (lt2 ? d : d2);
  int   ni2 = lt1 ? i1 : (lt2 ? t : i2);
  d0 = nd0; d1 = nd1; d2 = nd2;
  i0 = ni0; i1 = ni1; i2 = ni2;
}

// ---------------------------------------------------------------------------
// Kernel 1: per-triangle precompute. td[16] per (b,t):
//  [0..2]=d0 affine (A,B,C), [3..5]=d1, [6..8]=d2, [9..12]=bbox(minx,maxx,miny,maxy),
//  [13]=0, [14]=masked scaled area, [15]=0.
// bcol: per (b,t<NTRI_PAD) two float2: {-2cx,-2cy} and {cx^2+cy^2, 0} (WMMA A-rows).
// Also zeros sums4 bins and the 4 reduction accumulators.
// ---------------------------------------------------------------------------
__global__ void lv_prep(const float* __restrict__ gp, const int* __restrict__ tri,
                        const float* __restrict__ mask, float* __restrict__ td,
                        float* __restrict__ bcol, float* __restrict__ sums4,
                        float* __restrict__ acc) {
  int g = blockIdx.x * 256 + threadIdx.x;
  if (g < 4) acc[g] = 0.0f;
  if (g >= NB * NTRI_PAD) return;
  int b = g / NTRI_PAD, t = g % NTRI_PAD;
  float2* bc = ((float2*)bcol) + (size_t)(b * NTRI_PAD + t) * 2;
  if (t >= NTRI) {                       // padding -> huge distance sentinel
    bc[0] = make_float2(0.0f, 0.0f);
    bc[1] = make_float2(1e30f, 0.0f);
    return;
  }
  const int* tr = tri + ((size_t)b * NTRI + t) * 3;
  const float* gpb = gp + (size_t)b * NGP * 2;
  float v0x = gpb[tr[0] * 2], v0y = gpb[tr[0] * 2 + 1];
  float v1x = gpb[tr[1] * 2], v1y = gpb[tr[1] * 2 + 1];
  float v2x = gpb[tr[2] * 2], v2y = gpb[tr[2] * 2 + 1];

  // center -> WMMA A-row data
  float cx = (v0x + v1x + v2x) * (1.0f / 3.0f);
  float cy = (v0y + v1y + v2y) * (1.0f / 3.0f);
  bc[0] = make_float2(-2.0f * cx, -2.0f * cy);
  bc[1] = make_float2(cx * cx + cy * cy, 0.0f);

  float* e = td + ((size_t)b * NTRI + t) * 16;
  // edge sign affine form: d = -ey*px + ex*py + (ey*ax - ex*ay)
  float ex, ey;
  ex = v1x - v0x; ey = v1y - v0y;                  // d0: a=v0,b=v1
  e[0] = -ey; e[1] = ex; e[2] = ey * v0x - ex * v0y;
  ex = v2x - v1x; ey = v2y - v1y;                  // d1: a=v1,b=v2
  e[3] = -ey; e[4] = ex; e[5] = ey * v1x - ex * v1y;
  ex = v0x - v2x; ey = v0y - v2y;                  // d2: a=v2,b=v0
  e[6] = -ey; e[7] = ex; e[8] = ey * v2x - ex * v2y;
  e[9]  = fminf(fminf(v0x, v1x), v2x);
  e[10] = fmaxf(fmaxf(v0x, v1x), v2x);
  e[11] = fminf(fminf(v0y, v1y), v2y);
  e[12] = fmaxf(fmaxf(v0y, v1y), v2y);
  e[13] = 0.0f;
  // scaled signed area (scale = 20 on both coords)
  float ax = v0x * 20.0f, ay = v0y * 20.0f;
  float bx = v1x * 20.0f, by = v1y * 20.0f;
  float Cx = v2x * 20.0f, Cy = v2y * 20.0f;
  float area = ((ay + by) * (bx - ax) + (by + Cy) * (Cx - bx) + (Cy + ay) * (ax - Cx)) * 0.5f;
  e[14] = area * mask[b * NTRI + t];
  e[15] = 0.0f;

  float* s = sums4 + ((size_t)b * NTRI + t) * 4;
  s[0] = 0.0f; s[1] = 0.0f; s[2] = 0.0f; s[3] = 0.0f;
}

// ---------------------------------------------------------------------------
// Kernel 2: per-pixel first-hit triangle (condition) + f32 segment-sum atomics.
// Triangle table staged in LDS (25 KB); wave-uniform ds_load broadcasts.
// Early exit once every lane in the wave has found its first hit.
// ---------------------------------------------------------------------------
__global__ void lv_cond(const float* __restrict__ img, const float* __restrict__ td,
                        float* __restrict__ sums4, int* __restrict__ outCond) {
  __shared__ float4 s_td[NTRI * 4];
  int tid = threadIdx.x;
  int g = blockIdx.x * 256 + tid;             // [0, TOTPIX); 144 blocks per batch
  int b = g / NPIX, p = g - b * NPIX;
  const float4* tdb = (const float4*)(td + (size_t)b * NTRI * 16);
  for (int i = tid; i < NTRI * 4; i += 256) s_td[i] = tdb[i];
  __syncthreads();

  int x = p % IMG_W, y = p / IMG_W;
  float px = (x + 0.5f) * (1.0f / IMG_W);
  float py = (y + 0.5f) * (1.0f / IMG_H);
  int cond = 0;
  bool found = false;
  for (int t = 0; t < NTRI; ++t) {
    float4 e0 = s_td[t * 4 + 0];              // A0 B0 C0 A1
    float4 e1 = s_td[t * 4 + 1];              // B1 C1 A2 B2
    float4 e2 = s_td[t * 4 + 2];              // C2 minx maxx miny
    float4 e3 = s_td[t * 4 + 3];              // maxy . area .
    float d0 = fmaf(e0.x, px, fmaf(e0.y, py, e0.z));
    float d1 = fmaf(e0.w, px, fmaf(e1.x, py, e1.y));
    float d2 = fmaf(e1.z, px, fmaf(e1.w, py, e2.x));
    bool inside = ((d0 >= 0.0f) & (d1 >= 0.0f) & (d2 >= 0.0f)) |
                  ((d0 <= 0.0f) & (d1 <= 0.0f) & (d2 <= 0.0f));
    bool inb = (px >= e2.y) & (px <= e2.z) & (py >= e2.w) & (py <= e3.x);
    if (!found && inside && inb) { cond = t; found = true; }
    if (__all(found)) break;                  // wave-uniform early exit
  }
  outCond[g] = cond;
  float f0 = img[(size_t)g * 3 + 0];
  float f1 = img[(size_t)g * 3 + 1];
  float f2 = img[(size_t)g * 3 + 2];
  float* s = sums4 + ((size_t)b * NTRI + cond) * 4;
  atomicAdd(s + 0, f0);
  atomicAdd(s + 1, f1);
  atomicAdd(s + 2, f2);
  atomicAdd(s + 3, 1.0f);
}

// ---------------------------------------------------------------------------
// Kernel 3: WMMA distance matrix + per-pixel top-3 + softmax weights.
// One wave handles 16 pixels. M=16 triangles, N=16 pixels, K=4 per WMMA:
//   dist^2 = (px^2+py^2) + [-2cx,-2cy,cx^2+cy^2,0] . [px,py,1,0]
// All 25 A-tiles preloaded to VGPRs (25 x global_load_b64, one wait); the
// compute loop is fully unrolled: pure v_wmma + branchless cndmask selects,
// EXEC stays all-ones throughout (WMMA requirement).
// ---------------------------------------------------------------------------
__global__ void lv_topk(const float* __restrict__ bcol, float* __restrict__ idxw) {
  const int lane = threadIdx.x & 31;
  const int wave = (blockIdx.x * 256 + threadIdx.x) >> 5;
  const int pixbase = wave * 16;               // 16 pixels per wave, no batch straddle
  const int b = pixbase / NPIX;
  const int sub = lane & 15;
  const int gpix = pixbase + sub;
  const int p = gpix - b * NPIX;
  const float px = ((p % IMG_W) + 0.5f) * (1.0f / IMG_W);
  const float py = ((p / IMG_W) + 0.5f) * (1.0f / IMG_H);
  const float pp2 = px * px + py * py;

  v2f Bm;
  Bm.x = (lane < 16) ? px : 1.0f;              // K=0 / K=2
  Bm.y = (lane < 16) ? py : 0.0f;              // K=1 / K=3

  // preload all 25 A-tiles into registers (stride 16 tris * 2 float2 = 256 B)
  const float2* base = ((const float2*)bcol) +
                       (size_t)(b * NTRI_PAD + sub) * 2 + (lane >> 4);
  v2f Am[NTILE];
#pragma unroll
  for (int tile = 0; tile < NTILE; ++tile) {
    float2 q = base[tile * 32];
    Am[tile].x = q.x;
    Am[tile].y = q.y;
  }

  float d0 = 1e30f, d1 = 1e30f, d2 = 1e30f;
  int   i0 = 0,     i1 = 0,     i2 = 0;
  const int rofs = (lane >= 16) ? 8 : 0;

#pragma unroll
  for (int tile = 0; tile < NTILE; ++tile) {
    v8f c = {};
    c = __builtin_amdgcn_wmma_f32_16x16x4_f32(false, Am[tile], false, Bm,
                                              (short)0, c, false, false);
    const int tb = tile * 16 + rofs;
#pragma unroll
    for (int r = 0; r < 8; ++r) {
      top3_insert(c[r] + pp2, tb + r, d0, d1, d2, i0, i1, i2);
    }
  }
  // merge the two half-lists for this pixel (re-insert partner's sorted triple)
  float e0 = __shfl_xor(d0, 16, 32), e1 = __shfl_xor(d1, 16, 32), e2 = __shfl_xor(d2, 16, 32);
  int   j0 = __shfl_xor(i0, 16, 32), j1 = __shfl_xor(i1, 16, 32), j2 = __shfl_xor(i2, 16, 32);
  top3_insert(e0, j0, d0, d1, d2, i0, i1, i2);
  top3_insert(e1, j1, d0, d1, d2, i0, i1, i2);
  top3_insert(e2, j2, d0, d1, d2, i0, i1, i2);

  if (lane < 16) {
    const float inv_sigma = 14.0f / (0.001f * 20.0f);   // 1/sigma = 700
    float nd0 = -sqrtf(d0), nd1 = -sqrtf(d1), nd2 = -sqrtf(d2);
    float w0 = 1.0f;
    float w1 = __expf((nd1 - nd0) * inv_sigma);
    float w2 = __expf((nd2 - nd0) * inv_sigma);
    float inv = 1.0f / (w0 + w1 + w2);
    float* o = idxw + (size_t)gpix * 6;
    ((int*)o)[0] = i0; ((int*)o)[1] = i1; ((int*)o)[2] = i2;
    o[3] = w0 * inv; o[4] = w1 * inv; o[5] = w2 * inv;
  }
}

// ---------------------------------------------------------------------------
// Kernel 4: gather segment means, reconstruct, variance/L1 block reductions.
// ---------------------------------------------------------------------------
__global__ void lv_finish(const float* __restrict__ img, const float* __restrict__ sums4,
                          const float* __restrict__ idxw, float* __restrict__ recon,
                          float* __restrict__ acc) {
  __shared__ float sv[256];
  __shared__ float sl[256];
  int tid = threadIdx.x;
  int g = blockIdx.x * 256 + tid;
  int b = g / NPIX;
  const float* iw = idxw + (size_t)g * 6;
  int k0 = ((const int*)iw)[0], k1 = ((const int*)iw)[1], k2 = ((const int*)iw)[2];
  float w0 = iw[3], w1 = iw[4], w2 = iw[5];
  float f0 = img[(size_t)g * 3 + 0];
  float f1 = img[(size_t)g * 3 + 1];
  float f2 = img[(size_t)g * 3 + 2];

  const float* sb = sums4 + (size_t)b * NTRI * 4;
  float4 s0 = *(const float4*)(sb + (size_t)k0 * 4);
  float4 s1 = *(const float4*)(sb + (size_t)k1 * 4);
  float4 s2 = *(const float4*)(sb + (size_t)k2 * 4);
  float n0 = 1.0f / fmaxf(s0.w, 1.0f);
  float n1 = 1.0f / fmaxf(s1.w, 1.0f);
  float n2 = 1.0f / fmaxf(s2.w, 1.0f);
  float g00 = s0.x * n0, g01 = s0.y * n0, g02 = s0.z * n0;
  float g10 = s1.x * n1, g11 = s1.y * n1, g12 = s1.z * n1;
  float g20 = s2.x * n2, g21 = s2.y * n2, g22 = s2.z * n2;

  float r0 = w0 * g00 + w1 * g10 + w2 * g20;
  float r1 = w0 * g01 + w1 * g11 + w2 * g21;
  float r2 = w0 * g02 + w1 * g12 + w2 * g22;
  recon[(size_t)g * 3 + 0] = r0;
  recon[(size_t)g * 3 + 1] = r1;
  recon[(size_t)g * 3 + 2] = r2;

  float q0 = (f0 - g00) * (f0 - g00) + (f1 - g01) * (f1 - g01) + (f2 - g02) * (f2 - g02);
  float q1 = (f0 - g10) * (f0 - g10) + (f1 - g11) * (f1 - g11) + (f2 - g12) * (f2 - g12);
  float q2 = (f0 - g20) * (f0 - g20) + (f1 - g21) * (f1 - g21) + (f2 - g22) * (f2 - g22);
  float var_p = w0 * q0 + w1 * q1 + w2 * q2;
  float l1 = fabsf(r0 - f0) + fabsf(r1 - f1) + fabsf(r2 - f2);

  sv[tid] = var_p;
  sl[tid] = l1;
  __syncthreads();
  for (int o = 128; o > 0; o >>= 1) {
    if (tid < o) { sv[tid] += sv[tid + o]; sl[tid] += sl[tid + o]; }
    __syncthreads();
  }
  if (tid == 0) {
    atomicAdd(&acc[b], sv[0]);
    atomicAdd(&acc[2 + b], sl[0]);
  }
}

// ---------------------------------------------------------------------------
// Kernel 5: area variance (ddof=1) + final scalings of the reductions.
// ---------------------------------------------------------------------------
__global__ void lv_final(const float* __restrict__ td, const float* __restrict__ acc,
                         float* __restrict__ out) {
  __shared__ float ss[256];
  __shared__ float sq[256];
  int b = blockIdx.x, tid = threadIdx.x;
  float s = 0.0f, q = 0.0f;
  for (int t = tid; t < NTRI; t += 256) {
    float a = td[((size_t)b * NTRI + t) * 16 + 14];
    s += a; q += a * a;
  }
  ss[tid] = s; sq[tid] = q;
  __syncthreads();
  for (int o = 128; o > 0; o >>= 1) {
    if (tid < o) { ss[tid] += ss[tid + o]; sq[tid] += sq[tid + o]; }
    __syncthreads();
  }
  if (tid == 0) {
    float sumA = ss[0], sumQ = sq[0];
    float mean = sumA * (1.0f / NTRI);
    out[2 + b] = (sumQ - sumA * mean) * (1.0f / (NTRI - 1));   // area_variance
    out[0 + b] = acc[b] * (1.0f / (float)NPIX);                // variance
    out[4 + b] = acc[2 + b] * (1.0f / (float)(NPIX * 3));      // reconstruct_loss
  }
}

// ---------------------------------------------------------------------------
extern "C" void kernel_launch(void* const* d_in, const int* in_sizes, int n_in,
                              void* d_out, int out_size, void* d_ws, size_t ws_size,
                              hipStream_t stream) {
  (void)in_sizes; (void)n_in; (void)out_size; (void)ws_size;
  const float* gp   = (const float*)d_in[0];   // (B,225,2)
  const float* img  = (const float*)d_in[1];   // (B,H,W,3)
  const int*   tri  = (const int*)d_in[2];     // (B,392,3)
  const float* mask = (const float*)d_in[3];   // (B,392)

  float* w = (float*)d_ws;
  float* td    = w;            // 2*392*16 = 12544 floats
  float* bcol  = w + 12544;    // 2*400*4  =  3200 floats
  float* sums4 = w + 15744;    // 2*392*4  =  3136 floats
  float* acc   = w + 18880;    // 4 floats
  float* idxw  = w + 18884;    // 73728*6 floats

  float* o = (float*)d_out;
  float* recon   = o + 6;                   // (B,P,3) = 221184 floats
  int*   outCond = (int*)(o + 6 + 221184);  // (B,P) int32

  lv_prep  <<<(NB * NTRI_PAD + 255) / 256, 256, 0, stream>>>(gp, tri, mask, td, bcol, sums4, acc);
  lv_cond  <<<TOTPIX / 256, 256, 0, stream>>>(img, td, sums4, outCond);
  lv_topk  <<<(TOTPIX / 16) * 32 / 256, 256, 0, stream>>>(bcol, idxw);
  lv_finish<<<TOTPIX / 256, 256, 0, stream>>>(img, sums4, idxw, recon, acc);
  lv_final <<<NB, 256, 0, stream>>>(td, acc, o);
}